// PatchEmbed_HS_13073880449625
// MI455X (gfx1250) — compile-verified
//
#include <hip/hip_runtime.h>

typedef float v2f __attribute__((ext_vector_type(2)));
typedef float v8f __attribute__((ext_vector_type(8)));

// Reference: x:(256,3,224,224) f32, curve:(1024,2) i32, W:(384,3) f32, b:(384,) f32
//         -> out:(256,1,1024,384) f32
constexpr int Bn = 256;
constexpr int Cn = 3;
constexpr int Hn = 224;
constexpr int Wn = 224;
constexpr int Pn = 1024;
constexpr int En = 384;
constexpr int PTILES = Pn / 16;        // 64 point tiles of 16
constexpr int ETILES = En / 16;        // 24 embed tiles of 16
constexpr int WAVES_PER_BLOCK = 8;     // 256 threads, wave32

__global__ __launch_bounds__(256)
void patch_embed_wmma_f32(const float* __restrict__ x,
                          const int*   __restrict__ curve,
                          const float* __restrict__ Wm,
                          const float* __restrict__ bias,
                          float*       __restrict__ out)
{
    const int lane = threadIdx.x & 31;
    const int wave = threadIdx.x >> 5;
    const int gid  = blockIdx.x * WAVES_PER_BLOCK + wave;   // 0 .. 16383
    const int b    = gid >> 6;        // gid / PTILES
    const int pt   = gid & 63;        // gid % PTILES
    const int p0   = pt * 16;

    const int  m  = lane & 15;           // matrix row/col (M or N) owned by this lane
    const bool hi = (lane >= 16);        // lane group: K={0,1} (lo) vs K={2,3} (hi)
    const float lomask = hi ? 0.0f : 1.0f;   // zeroes the K=3 pad slot, branch-free

    // ---- Gather A tile: 16 points x 4 channels (channel 3 zero-padded) ----
    // One b64 load for the coordinate pair: curve[p] = {x, y}
    const int2 cv = ((const int2*)curve)[p0 + m];

    const float* xb = x + (size_t)b * Cn * Hn * Wn + (size_t)cv.y * Wn + cv.x;
    // ISA 32-bit A 16x4 layout: lanes 0-15: VGPR0=K0, VGPR1=K1; lanes 16-31: VGPR0=K2, VGPR1=K3
    // All loads unconditional from valid addresses; pad killed by mask multiply.
    const float xc01 = xb[0];                // c=0 (used by lo lanes)
    const float xc1  = xb[Hn * Wn];          // c=1 (valid for every lane)
    const float xc2  = xb[2 * Hn * Wn];      // c=2 (used by hi lanes)
    v2f a;
    a.x = hi ? xc2 : xc01;                   // v_cndmask, no branch
    a.y = xc1 * lomask;                      // K=3 pad -> 0 on hi lanes

    // Output base: VGPR r -> row p0 + r (lo lanes) / p0 + 8 + r (hi lanes), col m
    float* ob = out + ((size_t)b * Pn + p0 + (hi ? 8 : 0)) * En + m;

    for (int et = 0; et < ETILES; ++et) {
        const int e = et * 16 + m;           // embed column owned by this lane

        // ---- B tile: 4x16 slice of W^T (K x N), mirrored layout of A ----
        // Unconditional loads; select/mask in VALU only.
        const float w0 = Wm[e * 3 + 0];
        const float w1 = Wm[e * 3 + 1];
        const float w2 = Wm[e * 3 + 2];
        v2f bm;
        bm.x = hi ? w2 : w0;                 // v_cndmask
        bm.y = w1 * lomask;                  // K=3 pad -> 0 on hi lanes

        // ---- C accumulator pre-loaded with bias: C[m,n] = bias[e] for all m ----
        const float bv = bias[e];
        v8f c;
        #pragma unroll
        for (int i = 0; i < 8; ++i) c[i] = bv;

        // D = A x B + C  (16x16 f32 tile; bias folded into accumulator)
        v8f d = __builtin_amdgcn_wmma_f32_16x16x4_f32(
            /*neg_a=*/false, a, /*neg_b=*/false, bm,
            /*c_mod=*/(short)0, c, /*reuse_a=*/false, /*reuse_b=*/false);

        // ---- Store: 403 MB of output > 192 MB L2 -> stream non-temporal ----
        float* o = ob + et * 16;
        #pragma unroll
        for (int r = 0; r < 8; ++r)
            __builtin_nontemporal_store(d[r], o + (size_t)r * En);
    }
}

extern "C" void kernel_launch(void* const* d_in, const int* in_sizes, int n_in,
                              void* d_out, int out_size, void* d_ws, size_t ws_size,
                              hipStream_t stream)
{
    const float* x     = (const float*)d_in[0];
    const int*   curve = (const int*)  d_in[1];
    const float* Wm    = (const float*)d_in[2];
    const float* bias  = (const float*)d_in[3];
    float*       out   = (float*)d_out;

    const int total_waves = Bn * PTILES;                 // 16384
    const int blocks = total_waves / WAVES_PER_BLOCK;    // 2048
    patch_embed_wmma_f32<<<blocks, WAVES_PER_BLOCK * 32, 0, stream>>>(
        x, curve, Wm, bias, out);
}